// find_k_nearest_neighbors_25881472926490
// MI455X (gfx1250) — compile-verified
//
#include <hip/hip_runtime.h>
#include <hip/hip_bf16.h>

// Problem constants (match reference)
#define BB      16
#define CC      8
#define NSTA    512
#define LL      48
#define MGRID   16384          // LAT*LON
#define KNN     8
#define CHUNKS  32             // grid split into 32 chunks of 512 points
#define TILES_PER_CHUNK 32     // 32 tiles * 16 = 512 grid points per chunk
#define F4_PER_ROW 12          // L=48 floats = 12 float4
#define ROWS_PER_TENSOR ((long long)BB * CC * NSTA * KNN)       // 524288
#define U_PER_TENSOR (ROWS_PER_TENSOR * F4_PER_ROW)             // 6291456 float4

typedef __attribute__((ext_vector_type(2))) float v2f;
typedef __attribute__((ext_vector_type(8))) float v8f;

// Branchless sorted-ascending insertion of (x, xi) into top-8 register lists.
__device__ __forceinline__ void insert8(float x, int xi, float dk[KNN], int ik[KNN]) {
#pragma unroll
    for (int q = 0; q < KNN; ++q) {
        bool take = x < dk[q];
        float od = dk[q]; int oi = ik[q];
        dk[q] = take ? x  : dk[q];
        ik[q] = take ? xi : ik[q];
        x  = take ? od : x;
        xi = take ? oi : xi;
    }
}

// ---------------------------------------------------------------------------
// Kernel 1: per-(station-tile, grid-chunk) partial top-8 via WMMA distances.
//   d2[s][g] = |cobs_s - cera_g|^2 = A_s . B_g with
//   A_s = [-2x_s, -2y_s, x_s^2+y_s^2, 1],  B_g = [x_g, y_g, 1, x_g^2+y_g^2]
// One wave32 per block; V_WMMA_F32_16X16X4_F32 computes a 16x16 d2 tile.
// A-matrix f32 16x4 layout: lanes 0-15 hold K=0,1 (v0,v1) for M=lane;
//                           lanes 16-31 hold K=2,3 for M=lane-16.
// B-matrix 4x16 mirrors: lanes 0-15 hold rows K=0,1 at N=lane;
//                        lanes 16-31 hold rows K=2,3 at N=lane-16.
// ---------------------------------------------------------------------------
__global__ __launch_bounds__(32)
void knn_wmma_partial(const float* __restrict__ cobs,
                      const float* __restrict__ cera,
                      float* __restrict__ part_d,
                      int*   __restrict__ part_i)
{
    __shared__ float tile[16][17];
    const int lane  = threadIdx.x;
    const int s0    = blockIdx.x * 16;
    const int chunk = blockIdx.y;
    const int m     = lane & 15;
    const bool hi   = lane >= 16;

    // Loop-invariant A operand (stations)
    float sx = cobs[(s0 + m) * 2 + 0];
    float sy = cobs[(s0 + m) * 2 + 1];
    v2f a;
    if (!hi) { a.x = -2.0f * sx;       a.y = -2.0f * sy; }
    else     { a.x = sx * sx + sy * sy; a.y = 1.0f;      }

    float dk[KNN]; int ik[KNN];
#pragma unroll
    for (int q = 0; q < KNN; ++q) { dk[q] = 3.0e38f; ik[q] = 0; }

    const int gbase = chunk * (TILES_PER_CHUNK * 16);
    for (int t = 0; t < TILES_PER_CHUNK; ++t) {
        const int g0 = gbase + t * 16;
        const int g  = g0 + m;
        float gx = cera[g * 2 + 0];
        float gy = cera[g * 2 + 1];
        v2f b;
        if (!hi) { b.x = gx;   b.y = gy; }
        else     { b.x = 1.0f; b.y = gx * gx + gy * gy; }

        v8f c = {};
        // D = A x B (+0): emits v_wmma_f32_16x16x4_f32
        c = __builtin_amdgcn_wmma_f32_16x16x4_f32(
                /*neg_a=*/false, a, /*neg_b=*/false, b,
                /*c_mod=*/(short)0, c, /*reuse_a=*/false, /*reuse_b=*/false);

        __syncthreads();                       // protect tile vs prior reads
        const int mbase = hi ? 8 : 0;          // C/D layout: vgpr v -> row mbase+v, col m
#pragma unroll
        for (int v = 0; v < 8; ++v) tile[mbase + v][m] = c[v];
        __syncthreads();

        if (lane < 16) {                       // lane == station row of this tile
#pragma unroll
            for (int j = 0; j < 16; ++j) {
                insert8(tile[lane][j], g0 + j, dk, ik);
            }
        }
    }

    if (lane < 16) {
        const int s = s0 + lane;
        float* pd = part_d + ((size_t)s * CHUNKS + chunk) * KNN;
        int*   pi = part_i + ((size_t)s * CHUNKS + chunk) * KNN;
#pragma unroll
        for (int q = 0; q < KNN; ++q) { pd[q] = dk[q]; pi[q] = ik[q]; }
    }
}

// ---------------------------------------------------------------------------
// Kernel 2: merge 32 partial top-8 lists per station -> final top-8 indices.
// Also emits cera_k output (N*K*2 floats). One thread per station.
// ---------------------------------------------------------------------------
__global__ __launch_bounds__(128)
void knn_merge(const float* __restrict__ part_d,
               const int*   __restrict__ part_i,
               const float* __restrict__ cera,
               int*   __restrict__ idx_out,
               float* __restrict__ cera_k)
{
    int s = blockIdx.x * blockDim.x + threadIdx.x;
    if (s >= NSTA) return;

    float dk[KNN]; int ik[KNN];
#pragma unroll
    for (int q = 0; q < KNN; ++q) { dk[q] = 3.0e38f; ik[q] = 0; }

    const float* pd = part_d + (size_t)s * CHUNKS * KNN;
    const int*   pi = part_i + (size_t)s * CHUNKS * KNN;
    for (int r = 0; r < CHUNKS * KNN; ++r) {
        insert8(pd[r], pi[r], dk, ik);
    }

#pragma unroll
    for (int q = 0; q < KNN; ++q) {
        int g = ik[q];
        idx_out[s * KNN + q] = g;
        cera_k[(s * KNN + q) * 2 + 0] = cera[g * 2 + 0];
        cera_k[(s * KNN + q) * 2 + 1] = cera[g * 2 + 1];
    }
}

// ---------------------------------------------------------------------------
// Kernel 3: streaming gather, float4 granularity (b128 loads/stores).
// out float4 index t in [0, 2U): tensor = t/U (0=era,1=pan), r = t%U,
// row = r/12 encodes ((b*C+c)*N + s)*K + j, v = r%12.
// src float4 index = (bc*MGRID + idx[s*K+j])*12 + v. 12 consecutive lanes
// cover one 192B row -> coalesced reads and writes.
// ---------------------------------------------------------------------------
__global__ __launch_bounds__(256)
void knn_gather(const float* __restrict__ era,
                const float* __restrict__ pan,
                const int*   __restrict__ idx,
                float* __restrict__ out)
{
    unsigned int t = blockIdx.x * 256u + threadIdx.x;
    const unsigned int U = (unsigned int)U_PER_TENSOR;

    const float* src = era;
    unsigned int r = t;
    if (t >= U) { src = pan; r = t - U; }

    unsigned int v   = r % F4_PER_ROW;
    unsigned int row = r / F4_PER_ROW;                 // ((b*C+c)*N + s)*K + j
    unsigned int sj  = row % (NSTA * KNN);
    unsigned int bc  = row / (NSTA * KNN);

    int g = idx[sj];
    const float4* sp = (const float4*)src + ((size_t)bc * MGRID + (unsigned)g) * F4_PER_ROW + v;
    float4 d = *sp;
    ((float4*)out)[t] = d;
}

extern "C" void kernel_launch(void* const* d_in, const int* in_sizes, int n_in,
                              void* d_out, int out_size, void* d_ws, size_t ws_size,
                              hipStream_t stream) {
    // Input order: obs_his(unused), era_his, pan_fut, cobs, cera, cpan(unused), k
    const float* era  = (const float*)d_in[1];
    const float* pan  = (const float*)d_in[2];
    const float* cobs = (const float*)d_in[3];
    const float* cera = (const float*)d_in[4];

    float* out = (float*)d_out;

    // Workspace layout
    char* ws = (char*)d_ws;
    float* part_d = (float*)ws;                                   // 512*32*8*4  = 512 KB
    int*   part_i = (int*)(ws + (size_t)NSTA * CHUNKS * KNN * 4); // 512 KB
    int*   idx    = (int*)(ws + (size_t)NSTA * CHUNKS * KNN * 8); // 16 KB

    // Output layout: era_k | pan_k | cera_k
    const long long era_elems = ROWS_PER_TENSOR * LL;             // 25165824
    float* cera_k = out + 2 * era_elems;                          // 8192 floats

    // 1) WMMA distance tiles + per-chunk top-8 (32 station-tiles x 32 chunks)
    dim3 g1(NSTA / 16, CHUNKS);
    knn_wmma_partial<<<g1, 32, 0, stream>>>(cobs, cera, part_d, part_i);

    // 2) Merge partials -> final indices + cera_k
    knn_merge<<<(NSTA + 127) / 128, 128, 0, stream>>>(part_d, part_i, cera, idx, cera_k);

    // 3) Bandwidth-bound gather of era_k and pan_k (float4 streaming)
    unsigned int total_f4 = (unsigned int)(2 * U_PER_TENSOR);     // 12582912
    knn_gather<<<total_f4 / 256, 256, 0, stream>>>(era, pan, idx, out);
}